// MobileMambaBlock_43963285242157
// MI455X (gfx1250) — compile-verified
//
#include <hip/hip_runtime.h>
#include <hip/hip_bf16.h>
#include <math.h>

// ---------------------------------------------------------------------------
// MobileMamba block for MI455X (gfx1250, wave32).
// GEMMs use V_WMMA_F32_16X16X4_F32 with 32x32 register tiling per wave
// (4 WMMAs per 4 b64 loads -> 2x arithmetic intensity vs 16x16 tiles).
// ---------------------------------------------------------------------------

typedef float v2f __attribute__((ext_vector_type(2)));
typedef float v8f __attribute__((ext_vector_type(8)));

#define EPSF 1e-5f

#define WMMA_F32(av, bv, acc) \
  __builtin_amdgcn_wmma_f32_16x16x4_f32(false, (av), false, (bv), (short)0, (acc), false, false)

// ---------------- block reduction helper (non-pow2 blockDim) ---------------
__device__ __forceinline__ float block_reduce_sum(float v, float* buf, int n, int pow2) {
  int t = threadIdx.x;
  buf[t] = v;
  for (int q = t + n; q < pow2; q += n) buf[q] = 0.f;
  __syncthreads();
  for (int st = pow2 >> 1; st > 0; st >>= 1) {
    if (t < st) buf[t] += buf[t + st];
    __syncthreads();
  }
  float r = buf[0];
  __syncthreads();
  return r;
}

// ---------------- 1) Haar wavelet decomposition ----------------------------
// x (4,320,64,64) -> ll_tok (4096 tok x 192 ch, token-major), hi (4,576,32,32)
__global__ void k_wavelet(const float* __restrict__ x, float* __restrict__ ll_tok,
                          float* __restrict__ hi, int total) {
  int idx = blockIdx.x * blockDim.x + threadIdx.x;
  if (idx >= total) return;               // total = 4*192*1024
  int p = idx & 1023;
  int c = (idx >> 10) % 192;
  int b = idx / (192 << 10);
  int i = p >> 5, j = p & 31;
  const float* src = x + (((size_t)(b * 320 + c) * 64 + (i << 1)) << 6) + (j << 1);
  float a = src[0], bb = src[1], cc = src[64], dd = src[65];
  ll_tok[(size_t)((b << 10) + p) * 192 + c] = 0.5f * (a + bb + cc + dd);
  size_t hb = (size_t)(b * 576 + c * 3) << 10;
  hi[hb + p]        = 0.5f * (a + bb - cc - dd);   // f=1 (dhi x dlo)
  hi[hb + 1024 + p] = 0.5f * (a - bb + cc - dd);   // f=2 (dlo x dhi)
  hi[hb + 2048 + p] = 0.5f * (a - bb - cc + dd);   // f=3 (dhi x dhi)
}

// ---------------- 2) LayerNorm over 192 channels per token (in place) ------
__global__ void k_ln192(float* __restrict__ tok, const float* __restrict__ g,
                        const float* __restrict__ be) {
  __shared__ float buf[256];
  int t = threadIdx.x;
  size_t base = (size_t)blockIdx.x * 192;
  float v = tok[base + t];
  float s  = block_reduce_sum(v, buf, 192, 256);
  float s2 = block_reduce_sum(v * v, buf, 192, 256);
  float mean = s * (1.f / 192.f);
  float var  = s2 * (1.f / 192.f) - mean * mean;
  tok[base + t] = (v - mean) * rsqrtf(var + EPSF) * g[t] + be[t];
}

// ---------------- 3) GEMM  C[M,N] = A[M,K] * W[N,K]^T  (fp32 WMMA) ----------
// One wave32 computes a 32x32 tile: 2 A fragments x 2 B fragments per K-step
// -> 4 V_WMMA_F32_16X16X4_F32 per 4 global_load_b64 (2x intensity of 16x16).
// A/B lane layout per ISA: lanes 0-15 (row=lane, K=k..k+1),
// lanes 16-31 (row=lane-16, K=k+2..k+3). C/D: vgpr r -> row +8*half + r.
__global__ void k_gemm_wmma(const float* __restrict__ A, const float* __restrict__ W,
                            float* __restrict__ C, int M, int N, int K) {
  int wave = blockIdx.x * (blockDim.x >> 5) + (threadIdx.x >> 5);
  int lane = threadIdx.x & 31;
  int tilesN = N >> 5;
  int nTiles = (M >> 5) * tilesN;
  if (wave >= nTiles) return;
  int tm = (wave / tilesN) << 5;
  int tn = (wave % tilesN) << 5;
  int half = lane >> 4;
  int l = lane & 15;
  const float* a0 = A + (size_t)(tm + l) * K + (half << 1);
  const float* a1 = a0 + (size_t)16 * K;
  const float* b0 = W + (size_t)(tn + l) * K + (half << 1);
  const float* b1 = b0 + (size_t)16 * K;
  v8f acc00 = {}, acc01 = {}, acc10 = {}, acc11 = {};
#pragma unroll 2
  for (int k = 0; k < K; k += 4) {
    v2f av0 = *(const v2f*)(a0 + k);
    v2f av1 = *(const v2f*)(a1 + k);
    v2f bv0 = *(const v2f*)(b0 + k);
    v2f bv1 = *(const v2f*)(b1 + k);
    acc00 = WMMA_F32(av0, bv0, acc00);
    acc01 = WMMA_F32(av0, bv1, acc01);
    acc10 = WMMA_F32(av1, bv0, acc10);
    acc11 = WMMA_F32(av1, bv1, acc11);
  }
  float* c00 = C + (size_t)(tm + (half << 3)) * N + tn + l;
  float* c10 = c00 + (size_t)16 * N;
#pragma unroll
  for (int r = 0; r < 8; ++r) {
    c00[(size_t)r * N]      = acc00[r];
    c00[(size_t)r * N + 16] = acc01[r];
    c10[(size_t)r * N]      = acc10[r];
    c10[(size_t)r * N + 16] = acc11[r];
  }
}

// ---------------- 4) split xz -> xm channel-major ---------------------------
__global__ void k_split(const float* __restrict__ xz_tok, float* __restrict__ xm_chw,
                        int total) {
  int idx = blockIdx.x * blockDim.x + threadIdx.x;
  if (idx >= total) return;               // total = 4096*384
  int dd = idx % 384;
  int tok = idx / 384;
  int b = tok >> 10, p = tok & 1023;
  xm_chw[(((size_t)(b * 384 + dd)) << 10) + p] = xz_tok[(size_t)tok * 768 + dd];
}

// ---------------- 5) depthwise 3x3 (pad 1, 32x32) ---------------------------
// mode 0: +bias, SiLU, write token-major.  mode 1: BN+ReLU chmajor.  mode 2: BN.
__global__ void k_dwconv3(const float* __restrict__ in, const float* __restrict__ w,
                          const float* __restrict__ bias, const float* __restrict__ bnp,
                          float* __restrict__ out, int C, int mode, int total) {
  int idx = blockIdx.x * blockDim.x + threadIdx.x;
  if (idx >= total) return;
  int p = idx & 1023;
  int c = (idx >> 10) % C;
  int b = idx / (C << 10);
  int i = p >> 5, j = p & 31;
  const float* src = in + ((size_t)(b * C + c) << 10);
  const float* wc = w + c * 9;
  float acc = 0.f;
#pragma unroll
  for (int di = -1; di <= 1; ++di) {
    int ii = i + di;
    if (ii < 0 || ii > 31) continue;
#pragma unroll
    for (int dj = -1; dj <= 1; ++dj) {
      int jj = j + dj;
      if (jj < 0 || jj > 31) continue;
      acc += src[(ii << 5) + jj] * wc[(di + 1) * 3 + (dj + 1)];
    }
  }
  if (mode == 0) {
    acc += bias[c];
    acc = acc / (1.f + __expf(-acc));                       // SiLU
    out[(size_t)((b << 10) + p) * C + c] = acc;             // token-major
  } else {
    float g = bnp[c], be = bnp[C + c], m = bnp[2 * C + c], v = bnp[3 * C + c];
    acc = (acc - m) * (g * rsqrtf(v + EPSF)) + be;
    if (mode == 1) acc = fmaxf(acc, 0.f);
    out[((size_t)(b * C + c) << 10) + p] = acc;
  }
}

// ---------------- 6) x_proj + dt_proj + softplus per spatial position -------
// block = one (b,i,j); writes xdbl rows for all 4 scan directions and delta.
__global__ void k_xproj(const float* __restrict__ xconv_tok, const float* __restrict__ xpw,
                        const float* __restrict__ dtw, const float* __restrict__ dtb,
                        float* __restrict__ xdbl, float* __restrict__ delta) {
  __shared__ float sv[384];
  __shared__ float sdt[48];
  int blk = blockIdx.x;                   // b*1024 + p
  int b = blk >> 10, p = blk & 1023;
  int i = p >> 5, j = p & 31;
  int l0 = p, l1 = (j << 5) + i;
  int rowl[4] = {l0, l1, 1023 - l0, 1023 - l1};
  int t = threadIdx.x;
  for (int d = t; d < 384; d += 256) sv[d] = xconv_tok[(size_t)blk * 384 + d];
  __syncthreads();
  if (t < 176) {
    int k = t / 44, c = t % 44;
    const float* wrow = xpw + (size_t)(k * 44 + c) * 384;
    float acc = 0.f;
    for (int d = 0; d < 384; ++d) acc += sv[d] * wrow[d];
    int row = ((b * 4 + k) << 10) + rowl[k];
    xdbl[(size_t)row * 44 + c] = acc;
    if (c < 12) sdt[k * 12 + c] = acc;
  }
  __syncthreads();
  for (int q = t; q < 1536; q += 256) {
    int k = q / 384, d = q - k * 384;
    float acc = dtb[k * 384 + d];
    const float* wr = dtw + (size_t)(k * 384 + d) * 12;
#pragma unroll
    for (int r = 0; r < 12; ++r) acc += sdt[k * 12 + r] * wr[r];
    float sp = (acc > 20.f) ? acc : log1pf(__expf(acc));    // softplus
    int row = ((b * 4 + k) << 10) + rowl[k];
    delta[(size_t)row * 384 + d] = sp;
  }
}

// ---------------- 7) selective scan: one thread per (b,k,d) sequence --------
__global__ void k_scan(const float* __restrict__ delta, const float* __restrict__ xdbl,
                       const float* __restrict__ xconv_tok, const float* __restrict__ Alogs,
                       const float* __restrict__ Ds, float* __restrict__ ybuf) {
  int tid = blockIdx.x * blockDim.x + threadIdx.x;
  if (tid >= 6144) return;
  int d = tid % 384;
  int k = (tid / 384) & 3;
  int b = tid / 1536;
  float a[16];
#pragma unroll
  for (int n = 0; n < 16; ++n) a[n] = -__expf(Alogs[(size_t)(k * 384 + d) * 16 + n]);
  float Dv = Ds[k * 384 + d];
  float h[16];
#pragma unroll
  for (int n = 0; n < 16; ++n) h[n] = 0.f;
  int base = (b * 4 + k) << 10;
  for (int l = 0; l < 1024; ++l) {
    int row = base + l;
    float dl = delta[(size_t)row * 384 + d];
    int off;
    if (k == 0) off = l;
    else if (k == 1) off = ((l & 31) << 5) + (l >> 5);
    else if (k == 2) off = 1023 - l;
    else { int m2 = 1023 - l; off = ((m2 & 31) << 5) + (m2 >> 5); }
    float u = xconv_tok[(size_t)((b << 10) + off) * 384 + d];
    float du = dl * u;
    const float* bc = xdbl + (size_t)row * 44;   // [12..27]=B, [28..43]=C (broadcast)
    float y = 0.f;
#pragma unroll
    for (int n = 0; n < 16; ++n) {
      float e = __expf(dl * a[n]);
      h[n] = e * h[n] + du * bc[12 + n];
      y += h[n] * bc[28 + n];
    }
    ybuf[(size_t)row * 384 + d] = y + Dv * u;
  }
}

// ---------------- 8) combine 4 directions + LN(384) + *silu(z) --------------
__global__ void k_combine_ln(const float* __restrict__ ybuf, const float* __restrict__ xz_tok,
                             const float* __restrict__ g, const float* __restrict__ be,
                             float* __restrict__ yln) {
  __shared__ float buf[512];
  int token = blockIdx.x;                 // b*1024 + p
  int b = token >> 10, p = token & 1023;
  int i = p >> 5, j = p & 31;
  int l0 = p, l1 = (j << 5) + i;
  int d = threadIdx.x;
  size_t B0 = (size_t)(((b * 4 + 0) << 10) + l0) * 384;
  size_t B1 = (size_t)(((b * 4 + 1) << 10) + l1) * 384;
  size_t B2 = (size_t)(((b * 4 + 2) << 10) + (1023 - l0)) * 384;
  size_t B3 = (size_t)(((b * 4 + 3) << 10) + (1023 - l1)) * 384;
  float v = ybuf[B0 + d] + ybuf[B1 + d] + ybuf[B2 + d] + ybuf[B3 + d];
  float s  = block_reduce_sum(v, buf, 384, 512);
  float s2 = block_reduce_sum(v * v, buf, 384, 512);
  float mean = s * (1.f / 384.f);
  float var  = s2 * (1.f / 384.f) - mean * mean;
  float z = xz_tok[(size_t)token * 768 + 384 + d];
  float sz = z / (1.f + __expf(-z));
  yln[(size_t)token * 384 + d] =
      ((v - mean) * rsqrtf(var + EPSF) * g[d] + be[d]) * sz;
}

// ---------------- 9) SE: global pool / gate / apply -------------------------
__global__ void k_pool(const float* __restrict__ h2, float* __restrict__ s) {
  __shared__ float buf[256];
  int bc = blockIdx.x;                    // b*576 + c
  const float* src = h2 + ((size_t)bc << 10);
  float v = 0.f;
  for (int q = threadIdx.x; q < 1024; q += 256) v += src[q];
  buf[threadIdx.x] = v;
  __syncthreads();
  for (int st = 128; st > 0; st >>= 1) {
    if (threadIdx.x < st) buf[threadIdx.x] += buf[threadIdx.x + st];
    __syncthreads();
  }
  if (threadIdx.x == 0) s[bc] = buf[0] * (1.f / 1024.f);
}

__global__ void k_segate(const float* __restrict__ s, const float* __restrict__ rw,
                         const float* __restrict__ rb, const float* __restrict__ ew,
                         const float* __restrict__ eb, float* __restrict__ gate) {
  __shared__ float ss[576];
  __shared__ float rr[144];
  int b = blockIdx.x, t = threadIdx.x;
  ss[t] = s[b * 576 + t];
  __syncthreads();
  if (t < 144) {
    float acc = rb[t];
    const float* w = rw + (size_t)t * 576;
    for (int c = 0; c < 576; ++c) acc += ss[c] * w[c];
    rr[t] = fmaxf(acc, 0.f);
  }
  __syncthreads();
  float acc = eb[t];
  const float* w = ew + (size_t)t * 144;
  for (int r = 0; r < 144; ++r) acc += rr[r] * w[r];
  gate[b * 576 + t] = 1.f / (1.f + __expf(-acc));
}

__global__ void k_gapply(const float* __restrict__ h2, const float* __restrict__ gate,
                         float* __restrict__ h2tok, int total) {
  int idx = blockIdx.x * blockDim.x + threadIdx.x;
  if (idx >= total) return;               // total = 4*1024*576, token-major out
  int c = idx % 576;
  int tok = idx / 576;
  int b = tok >> 10, p = tok & 1023;
  h2tok[idx] = h2[(((size_t)(b * 576 + c)) << 10) + p] * gate[b * 576 + c];
}

// ---------------- 10) wavelet reconstruction -> xout tokens (320ch) ---------
__global__ void k_recon(const float* __restrict__ outll_tok, const float* __restrict__ pwtok,
                        const float* __restrict__ bn3, const float* __restrict__ x,
                        float* __restrict__ xout_tok, int total) {
  int idx = blockIdx.x * blockDim.x + threadIdx.x;
  if (idx >= total) return;               // total = 4*4096*320
  int c = idx % 320;
  int t = idx / 320;                      // b*4096 + y*64 + x
  int b = t >> 12;
  int yy = (t >> 6) & 63, xx = t & 63;
  float val;
  if (c < 192) {
    int i = yy >> 1, dd = yy & 1, j = xx >> 1, ee = xx & 1;
    int tok32 = (b << 10) + (i << 5) + j;
    float ll = outll_tok[(size_t)tok32 * 192 + c];
    float vf[3];
#pragma unroll
    for (int f = 0; f < 3; ++f) {
      int o = c * 3 + f;
      float g = bn3[o], be = bn3[576 + o], m = bn3[1152 + o], vv = bn3[1728 + o];
      vf[f] = (pwtok[(size_t)tok32 * 576 + o] - m) * (g * rsqrtf(vv + EPSF)) + be;
    }
    float s1 = (dd == 0) ? -1.f : 1.f;          // rec[1] = 0.5*[[-1,-1],[1,1]]
    float s2 = (ee == 0) ? -1.f : 1.f;          // rec[2] = 0.5*[[-1,1],[-1,1]]
    float s3 = (dd == ee) ? 1.f : -1.f;         // rec[3] = 0.5*[[1,-1],[-1,1]]
    val = 0.5f * (ll + s1 * vf[0] + s2 * vf[1] + s3 * vf[2]);
  } else {
    val = x[((size_t)(b * 320 + c) * 64 + yy) * 64 + xx];   // identity channels
  }
  xout_tok[idx] = val;
}

// ---------------- 11) final BN + residual -----------------------------------
__global__ void k_final(const float* __restrict__ projtok, const float* __restrict__ pbn,
                        const float* __restrict__ x, float* __restrict__ out, int total) {
  int idx = blockIdx.x * blockDim.x + threadIdx.x;
  if (idx >= total) return;               // (B,320,64,64)
  int xx = idx & 63;
  int yy = (idx >> 6) & 63;
  int c = (idx >> 12) % 320;
  int b = idx / (320 << 12);
  float pv = projtok[(size_t)((b << 12) + (yy << 6) + xx) * 320 + c];
  float g = pbn[c], be = pbn[320 + c], m = pbn[640 + c], v = pbn[960 + c];
  out[idx] = x[idx] + (pv - m) * (g * rsqrtf(v + EPSF)) + be;
}

// ---------------------------------------------------------------------------
extern "C" void kernel_launch(void* const* d_in, const int* in_sizes, int n_in,
                              void* d_out, int out_size, void* d_ws, size_t ws_size,
                              hipStream_t stream) {
  (void)in_sizes; (void)n_in; (void)out_size; (void)ws_size;
  const float* x        = (const float*)d_in[0];
  const float* inpw     = (const float*)d_in[1];
  const float* convw    = (const float*)d_in[2];
  const float* convb    = (const float*)d_in[3];
  const float* xpw      = (const float*)d_in[4];
  const float* dtw      = (const float*)d_in[5];
  const float* dtb      = (const float*)d_in[6];
  const float* alogs    = (const float*)d_in[7];
  const float* dsv      = (const float*)d_in[8];
  const float* ong      = (const float*)d_in[9];
  const float* onb      = (const float*)d_in[10];
  const float* outpw    = (const float*)d_in[11];
  const float* normg    = (const float*)d_in[12];
  const float* normb    = (const float*)d_in[13];
  const float* dw1w     = (const float*)d_in[14];
  const float* bn1      = (const float*)d_in[15];
  const float* dw2w     = (const float*)d_in[16];
  const float* bn2      = (const float*)d_in[17];
  const float* serw     = (const float*)d_in[18];
  const float* serb     = (const float*)d_in[19];
  const float* seew     = (const float*)d_in[20];
  const float* seeb     = (const float*)d_in[21];
  const float* pww      = (const float*)d_in[22];
  const float* bn3      = (const float*)d_in[23];
  const float* projw    = (const float*)d_in[24];
  const float* pbn      = (const float*)d_in[25];
  float* out = (float*)d_out;

  // ---- workspace carve-out ----
  size_t off = 0;
  char* base = (char*)d_ws;
  auto alloc = [&](size_t bytes) -> float* {
    float* p = (float*)(base + off);
    off = (off + bytes + 255) & ~(size_t)255;
    return p;
  };
  float* ll_tok   = alloc((size_t)4096 * 192 * 4);
  float* xz_tok   = alloc((size_t)4096 * 768 * 4);
  float* xm_chw   = alloc((size_t)4 * 384 * 1024 * 4);
  float* xconvtok = alloc((size_t)4096 * 384 * 4);
  float* xdbl     = alloc((size_t)16384 * 44 * 4);
  float* deltab   = alloc((size_t)16384 * 384 * 4);
  float* ybuf     = alloc((size_t)16384 * 384 * 4);
  float* yln      = alloc((size_t)4096 * 384 * 4);
  float* outll    = alloc((size_t)4096 * 192 * 4);
  float* hi       = alloc((size_t)4 * 576 * 1024 * 4);
  float* h1       = alloc((size_t)4 * 576 * 1024 * 4);
  float* h2       = alloc((size_t)4 * 576 * 1024 * 4);
  float* h2tok    = alloc((size_t)4 * 576 * 1024 * 4);
  float* pwtok    = alloc((size_t)4096 * 576 * 4);
  float* sbuf     = alloc((size_t)2304 * 4);
  float* gate     = alloc((size_t)2304 * 4);
  float* xouttok  = alloc((size_t)16384 * 320 * 4);
  float* projtok  = alloc((size_t)16384 * 320 * 4);

  // 1) wavelet decomposition
  {
    int total = 4 * 192 * 1024;
    k_wavelet<<<(total + 255) / 256, 256, 0, stream>>>(x, ll_tok, hi, total);
  }
  // 2) LN over 192 channels (token-major, in place)
  k_ln192<<<4096, 192, 0, stream>>>(ll_tok, normg, normb);
  // 3) in_proj GEMM: 4096x192 @ (768x192)^T -> xz_tok  (WMMA fp32, 32x32 tiles)
  {
    int tiles = (4096 / 32) * (768 / 32);
    k_gemm_wmma<<<(tiles + 7) / 8, 256, 0, stream>>>(ll_tok, inpw, xz_tok, 4096, 768, 192);
  }
  // 4) split xm to channel-major
  {
    int total = 4096 * 384;
    k_split<<<(total + 255) / 256, 256, 0, stream>>>(xz_tok, xm_chw, total);
  }
  // 5) dwconv3 + bias + SiLU -> token-major xconvtok
  {
    int total = 4 * 384 * 1024;
    k_dwconv3<<<(total + 255) / 256, 256, 0, stream>>>(xm_chw, convw, convb, nullptr,
                                                       xconvtok, 384, 0, total);
  }
  // 6) x_proj + dt_proj + softplus
  k_xproj<<<4096, 256, 0, stream>>>(xconvtok, xpw, dtw, dtb, xdbl, deltab);
  // 7) selective scan (6144 sequences, 1024 steps)
  k_scan<<<24, 256, 0, stream>>>(deltab, xdbl, xconvtok, alogs, dsv, ybuf);
  // 8) combine directions + LN + silu(z)
  k_combine_ln<<<4096, 384, 0, stream>>>(ybuf, xz_tok, ong, onb, yln);
  // 9) out_proj GEMM: 4096x384 @ (192x384)^T -> outll
  {
    int tiles = (4096 / 32) * (192 / 32);
    k_gemm_wmma<<<(tiles + 7) / 8, 256, 0, stream>>>(yln, outpw, outll, 4096, 192, 384);
  }
  // 10) HF path: dwconv+BN+ReLU, dwconv+BN
  {
    int total = 4 * 576 * 1024;
    k_dwconv3<<<(total + 255) / 256, 256, 0, stream>>>(hi, dw1w, nullptr, bn1, h1, 576, 1, total);
    k_dwconv3<<<(total + 255) / 256, 256, 0, stream>>>(h1, dw2w, nullptr, bn2, h2, 576, 2, total);
  }
  // 11) SE gate
  k_pool<<<4 * 576, 256, 0, stream>>>(h2, sbuf);
  k_segate<<<4, 576, 0, stream>>>(sbuf, serw, serb, seew, seeb, gate);
  {
    int total = 4 * 1024 * 576;
    k_gapply<<<(total + 255) / 256, 256, 0, stream>>>(h2, gate, h2tok, total);
  }
  // 12) pointwise 576->576 GEMM (WMMA)
  {
    int tiles = (4096 / 32) * (576 / 32);
    k_gemm_wmma<<<(tiles + 7) / 8, 256, 0, stream>>>(h2tok, pww, pwtok, 4096, 576, 576);
  }
  // 13) wavelet reconstruction + identity concat -> xouttok
  {
    int total = 4 * 4096 * 320;
    k_recon<<<(total + 255) / 256, 256, 0, stream>>>(outll, pwtok, bn3, x, xouttok, total);
  }
  // 14) proj 320->320 GEMM over 16384 positions (WMMA)
  {
    int tiles = (16384 / 32) * (320 / 32);
    k_gemm_wmma<<<(tiles + 7) / 8, 256, 0, stream>>>(xouttok, projw, projtok, 16384, 320, 320);
  }
  // 15) final BN + residual
  {
    int total = 4 * 320 * 64 * 64;
    k_final<<<(total + 255) / 256, 256, 0, stream>>>(projtok, pbn, x, out, total);
  }
}